// WeightedGINGraphAutoEncoder_24876450579019
// MI455X (gfx1250) — compile-verified
//
#include <hip/hip_runtime.h>
#include <hip/hip_bf16.h>
#include <math.h>

// ---------------------------------------------------------------------------
// WeightedGIN graph autoencoder for MI455X (gfx1250, wave32).
//   layer: agg = scatter_add(w * x[src] -> dst);  h = relu(agg@W1^T)@W2^T
//   two layers, then row-wise L2 normalize.
// fp32 end-to-end; GEMMs use V_WMMA_F32_16X16X4_F32 (CDNA5 wave-matrix).
// ---------------------------------------------------------------------------

typedef __attribute__((ext_vector_type(2))) float v2f;
typedef __attribute__((ext_vector_type(8))) float v8f;

#define DIM 64
#define LDS_STRIDE 68   // 64 + 4 pad: conflict-free column-ish reloads, 8B aligned

// ---------------------------------------------------------------- zero fill
__global__ __launch_bounds__(256) void gin_zero_kernel(float4* __restrict__ p, int n4) {
    int i = blockIdx.x * blockDim.x + threadIdx.x;
    if (i < n4) p[i] = float4{0.f, 0.f, 0.f, 0.f};
}

// ------------------------------------------------------------- scatter-add
// 16 threads per edge; each thread handles 4 consecutive feature dims.
__global__ __launch_bounds__(256) void gin_scatter_kernel(
    const float* __restrict__ xin, const int* __restrict__ src,
    const int* __restrict__ dst, const float* __restrict__ w,
    float* __restrict__ agg, int n_edges) {
    int t = blockIdx.x * blockDim.x + threadIdx.x;
    int e = t >> 4;
    if (e >= n_edges) return;
    int dg = (t & 15) * 4;
    int s = src[e];
    int d = dst[e];
    float ww = w[e];
    const float4 v = *(const float4*)(xin + (size_t)s * DIM + dg);
    float* ap = agg + (size_t)d * DIM + dg;
    atomicAdd(ap + 0, ww * v.x);
    atomicAdd(ap + 1, ww * v.y);
    atomicAdd(ap + 2, ww * v.z);
    atomicAdd(ap + 3, ww * v.w);
}

// ------------------------------------------------- fused MLP: relu(A@W1^T)@W2^T
// One wave (32 lanes) per 16-row tile. N=64 handled as 4 tiles of 16.
// A-layout (16x4 f32): lane = M (0..15), khalf = lane>>4 selects K pair:
//   a = { A[m][kk + 2*khalf], A[m][kk + 2*khalf + 1] }   (contiguous float2)
// B-layout mirrors: b = { B[kk+2*khalf][n], B[kk+2*khalf+1][n] } with
//   B[k][n] = W[n][k]  (torch Linear: y = x @ W^T)  -> contiguous float2 of W row n.
// C/D layout: VGPR r, lanes 0-15 -> M=r, lanes 16-31 -> M=r+8; N = lane&15.
__global__ __launch_bounds__(32) void gin_mlp_kernel(
    const float* __restrict__ agg, const float* __restrict__ W1,
    const float* __restrict__ W2, float* __restrict__ out, int n_nodes) {
    __shared__ float hs[16 * LDS_STRIDE];

    const int lane  = threadIdx.x;      // 0..31 (wave32)
    const int m     = lane & 15;
    const int khalf = lane >> 4;        // 0 or 1
    const int row0  = blockIdx.x * 16;

    int arow_idx = row0 + m;
    if (arow_idx >= n_nodes) arow_idx = n_nodes - 1;   // safe read; stores guarded
    const float* arow = agg + (size_t)arow_idx * DIM;

    // ---- GEMM1: H = AGG @ W1^T
    v8f acc[4];
    #pragma unroll
    for (int nt = 0; nt < 4; ++nt) {
        v8f c = {};
        const float* brow = W1 + (size_t)(nt * 16 + m) * DIM;
        #pragma unroll
        for (int kk = 0; kk < DIM; kk += 4) {
            const int k0 = kk + 2 * khalf;
            v2f a = *(const v2f*)(arow + k0);
            v2f b = *(const v2f*)(brow + k0);
            c = __builtin_amdgcn_wmma_f32_16x16x4_f32(
                    false, a, false, b, (short)0, c, false, false);
        }
        acc[nt] = c;
    }

    // ---- ReLU, stage H (16x64) in LDS in row-major A-order
    #pragma unroll
    for (int nt = 0; nt < 4; ++nt) {
        #pragma unroll
        for (int r = 0; r < 8; ++r) {
            float v = acc[nt][r];
            v = v > 0.0f ? v : 0.0f;
            const int mm = r + 8 * khalf;
            const int nn = nt * 16 + m;
            hs[mm * LDS_STRIDE + nn] = v;
        }
    }
    __syncthreads();

    // ---- GEMM2: OUT = H @ W2^T
    const float* hrow = hs + m * LDS_STRIDE;
    #pragma unroll
    for (int nt = 0; nt < 4; ++nt) {
        v8f c = {};
        const float* brow = W2 + (size_t)(nt * 16 + m) * DIM;
        #pragma unroll
        for (int kk = 0; kk < DIM; kk += 4) {
            const int k0 = kk + 2 * khalf;
            v2f a = *(const v2f*)(hrow + k0);
            v2f b = *(const v2f*)(brow + k0);
            c = __builtin_amdgcn_wmma_f32_16x16x4_f32(
                    false, a, false, b, (short)0, c, false, false);
        }
        #pragma unroll
        for (int r = 0; r < 8; ++r) {
            const int mm = r + 8 * khalf;
            const int nn = nt * 16 + m;
            if (row0 + mm < n_nodes)
                out[(size_t)(row0 + mm) * DIM + nn] = c[r];
        }
    }
}

// ------------------------------------------------------- row L2 normalize
// One wave per row; 2 floats per lane; wave32 xor-shuffle reduction. In-place.
__global__ __launch_bounds__(32) void gin_l2norm_kernel(float* __restrict__ xio) {
    const int row  = blockIdx.x;
    const int lane = threadIdx.x;
    float* p = xio + (size_t)row * DIM + lane * 2;
    float2 v = *(const float2*)p;
    float ss = v.x * v.x + v.y * v.y;
    #pragma unroll
    for (int off = 16; off > 0; off >>= 1)
        ss += __shfl_xor(ss, off, 32);
    const float inv = 1.0f / fmaxf(sqrtf(ss), 1e-12f);
    v.x *= inv; v.y *= inv;
    *(float2*)p = v;
}

// ---------------------------------------------------------------------------
extern "C" void kernel_launch(void* const* d_in, const int* in_sizes, int n_in,
                              void* d_out, int out_size, void* d_ws, size_t ws_size,
                              hipStream_t stream) {
    const float* x    = (const float*)d_in[0];   // [N, 64]
    const int*   ei   = (const int*)d_in[1];     // [2, E] (int32 on device)
    const float* ew   = (const float*)d_in[2];   // [E]
    const float* W1_0 = (const float*)d_in[3];   // [64, 64]
    const float* W2_0 = (const float*)d_in[4];
    const float* W1_1 = (const float*)d_in[5];
    const float* W2_1 = (const float*)d_in[6];
    float* out = (float*)d_out;

    const int n_nodes = in_sizes[0] / DIM;       // 50000
    const int n_edges = in_sizes[2];             // 800000
    const int* src = ei;                         // edge_index[0]
    const int* dst = ei + n_edges;               // edge_index[1]

    // workspace: agg [N,64] + h [N,64]  (25.6 MB)
    float* agg = (float*)d_ws;
    float* h   = agg + (size_t)n_nodes * DIM;

    const int n4 = (n_nodes * DIM) / 4;
    dim3 zgrid((n4 + 255) / 256);
    dim3 sgrid((unsigned)(((size_t)n_edges * 16 + 255) / 256));
    dim3 mgrid((n_nodes + 15) / 16);

    // ---- layer 0
    gin_zero_kernel<<<zgrid, 256, 0, stream>>>((float4*)agg, n4);
    gin_scatter_kernel<<<sgrid, 256, 0, stream>>>(x, src, dst, ew, agg, n_edges);
    gin_mlp_kernel<<<mgrid, 32, 0, stream>>>(agg, W1_0, W2_0, h, n_nodes);

    // ---- layer 1 (MLP writes straight to d_out)
    gin_zero_kernel<<<zgrid, 256, 0, stream>>>((float4*)agg, n4);
    gin_scatter_kernel<<<sgrid, 256, 0, stream>>>(h, src, dst, ew, agg, n_edges);
    gin_mlp_kernel<<<mgrid, 32, 0, stream>>>(agg, W1_1, W2_1, out, n_nodes);

    // ---- normalize rows in-place
    gin_l2norm_kernel<<<n_nodes, 32, 0, stream>>>(out);
}